// PhenoDrugModel_43645457662441
// MI455X (gfx1250) — compile-verified
//
#include <hip/hip_runtime.h>
#include <hip/hip_bf16.h>

#define NN 50000
#define DD 128
#define RR 16
#define NBASIS 10
#define EE 500000
#define BB 1024
#define PP 64
#define HH 4
#define DHH 32

typedef float v2f __attribute__((ext_vector_type(2)));
typedef float v8f __attribute__((ext_vector_type(8)));

// meta layout (ints): [0..15] typeCount, [16..32] typeOffset, [33..49] tileOffset, [50..65] cursor
#define MO_CNT 0
#define MO_OFF 16
#define MO_TILE 33
#define MO_CUR 50

// ---------------------------------------------------------------- histogram / sort
__global__ void k_type_count(const int* __restrict__ etype, int* __restrict__ meta) {
    int i = blockIdx.x * blockDim.x + threadIdx.x;
    if (i < EE) atomicAdd(&meta[MO_CNT + etype[i]], 1);
}

__global__ void k_scan(int* __restrict__ meta) {
    if (threadIdx.x == 0 && blockIdx.x == 0) {
        int off = 0, toff = 0;
        meta[MO_OFF + 0] = 0;
        meta[MO_TILE + 0] = 0;
        for (int r = 0; r < RR; ++r) {
            int c = meta[MO_CNT + r];
            meta[MO_CUR + r] = off;
            off += c;
            toff += (c + 15) / 16;
            meta[MO_OFF + r + 1] = off;
            meta[MO_TILE + r + 1] = toff;
        }
    }
}

__global__ void k_permute(const int* __restrict__ etype, int* __restrict__ meta,
                          int* __restrict__ perm) {
    int i = blockIdx.x * blockDim.x + threadIdx.x;
    if (i < EE) {
        int p = atomicAdd(&meta[MO_CUR + etype[i]], 1);
        perm[p] = i;
    }
}

__global__ void k_seg_count(const int* __restrict__ ei, const int* __restrict__ etype,
                            float* __restrict__ cnt) {
    int i = blockIdx.x * blockDim.x + threadIdx.x;
    if (i < EE) atomicAdd(&cnt[etype[i] * NN + ei[EE + i]], 1.0f);
}

__global__ void k_inv(float* __restrict__ cnt) {
    int i = blockIdx.x * blockDim.x + threadIdx.x;
    if (i < RR * NN) cnt[i] = 1.0f / fmaxf(cnt[i], 1.0f);
}

// ---------------------------------------------------------------- packed-W builders
// Packed fragment layout (per matrix): Wp[((nt*16 + k2)*32 + lane)*4 + j]
//   hi = lane>>4, mlo = lane&15, n = nt*16+mlo
//   j0/j1 <- W[8*k2+2*hi + 0/1][n]   (WMMA k4 = 2*k2)
//   j2/j3 <- W[8*k2+4+2*hi + 0/1][n] (WMMA k4 = 2*k2+1)
// One global_load_b128 per wave feeds TWO v_wmma_f32_16x16x4_f32.

// Fused basis combination + packing: Wp for all R relations.
__global__ void k_make_Wp(const float* __restrict__ comp, const float* __restrict__ basis,
                          float* __restrict__ Wp) {
    int idx = blockIdx.x * blockDim.x + threadIdx.x;   // R*8*16*32 = 65536
    if (idx >= RR * 8 * 16 * 32) return;
    int lane = idx & 31;
    int k2 = (idx >> 5) & 15;
    int nt = (idx >> 9) & 7;
    int r = idx >> 12;
    int hi = lane >> 4, mlo = lane & 15;
    int o = nt * 16 + mlo;
    int d0 = 8 * k2 + 2 * hi;
    int d2 = d0 + 4;
    float4 v = make_float4(0.f, 0.f, 0.f, 0.f);
    for (int b = 0; b < NBASIS; ++b) {
        float c = comp[r * NBASIS + b];
        const float* bs = basis + b * DD * DD;
        v.x += c * bs[d0 * DD + o];
        v.y += c * bs[(d0 + 1) * DD + o];
        v.z += c * bs[d2 * DD + o];
        v.w += c * bs[(d2 + 1) * DD + o];
    }
    *(float4*)(Wp + (size_t)idx * 4) = v;
}

// Pack a single DxD matrix (root) into fragment order.
__global__ void k_pack_root(const float* __restrict__ W, float* __restrict__ Wp) {
    int idx = blockIdx.x * blockDim.x + threadIdx.x;   // 8*16*32 = 4096
    if (idx >= 8 * 16 * 32) return;
    int lane = idx & 31;
    int k2 = (idx >> 5) & 15;
    int nt = (idx >> 9) & 7;
    int hi = lane >> 4, mlo = lane & 15;
    int o = nt * 16 + mlo;
    int d0 = 8 * k2 + 2 * hi;
    int d2 = d0 + 4;
    float4 v;
    v.x = W[d0 * DD + o];
    v.y = W[(d0 + 1) * DD + o];
    v.z = W[d2 * DD + o];
    v.w = W[(d2 + 1) * DD + o];
    *(float4*)(Wp + (size_t)idx * 4) = v;
}

// ---------------------------------------------------------------- root GEMM: h = x @ root + cbias
__global__ void __launch_bounds__(128) k_root(const float* __restrict__ x,
                                              const float* __restrict__ Rp,   // packed root
                                              const float* __restrict__ cbias,
                                              float* __restrict__ h) {
    __shared__ float lds[4][16][132];
    int wave = threadIdx.x >> 5, lane = threadIdx.x & 31;
    int mt = blockIdx.x * 4 + wave;
    if (mt >= NN / 16) return;
    // async-stage 16 rows of x into LDS (b128 per lane = full 512B row per wave)
    for (int i = 0; i < 16; ++i) {
        const float* gp = x + (size_t)(mt * 16 + i) * DD + lane * 4;
        unsigned loff = (unsigned)(size_t)&lds[wave][i][lane * 4];
        asm volatile("global_load_async_to_lds_b128 %0, %1, off"
                     :: "v"(loff), "v"(gp) : "memory");
    }
    asm volatile("s_wait_asynccnt 0x0" ::: "memory");
    int hi = lane >> 4, mlo = lane & 15;
    for (int nt = 0; nt < 8; ++nt) {
        v8f acc = {};
        const float* Bp = Rp + (size_t)nt * 16 * 128;
        for (int k2 = 0; k2 < 16; ++k2) {
            float4 bb = *(const float4*)(Bp + k2 * 128 + lane * 4);
            const float* arow = &lds[wave][mlo][8 * k2 + 2 * hi];
            v2f a0; a0.x = arow[0]; a0.y = arow[1];
            v2f b0; b0.x = bb.x; b0.y = bb.y;
            acc = __builtin_amdgcn_wmma_f32_16x16x4_f32(false, a0, false, b0,
                                                        (short)0, acc, false, false);
            v2f a1; a1.x = arow[4]; a1.y = arow[5];
            v2f b1; b1.x = bb.z; b1.y = bb.w;
            acc = __builtin_amdgcn_wmma_f32_16x16x4_f32(false, a1, false, b1,
                                                        (short)0, acc, false, false);
        }
        for (int v = 0; v < 8; ++v) {
            int row = mt * 16 + v + 8 * hi;
            int col = nt * 16 + mlo;
            h[row * DD + col] = acc[v] + cbias[col];
        }
    }
}

// ---------------------------------------------------------------- edge scatter GEMM
__global__ void __launch_bounds__(128) k_scatter(const int* __restrict__ ei,
                                                 const float* __restrict__ x,
                                                 const float* __restrict__ Wp,
                                                 const float* __restrict__ inv,
                                                 const int* __restrict__ perm,
                                                 const int* __restrict__ meta,
                                                 float* __restrict__ h) {
    __shared__ float lds[4][16][132];
    __shared__ float sInv[4][16];
    __shared__ int sDst[4][16];
    __shared__ int sSrc[4][16];
    int wave = threadIdx.x >> 5, lane = threadIdx.x & 31;
    int tileId = blockIdx.x * 4 + wave;
    int totalTiles = meta[MO_TILE + RR];
    if (tileId >= totalTiles) return;
    // find relation r for this tile
    int r = 0;
    for (int i = 1; i < RR; ++i)
        if (tileId >= meta[MO_TILE + i]) r = i;
    int localT = tileId - meta[MO_TILE + r];
    int base = meta[MO_OFF + r] + localT * 16;
    int count = meta[MO_OFF + r + 1] - meta[MO_OFF + r];
    int nvalid = min(16, count - localT * 16);
    if (lane < 16) {
        int dstv = -1, srcv = 0;   // invalid rows: read row 0 (harmless; output masked)
        float iv = 0.f;
        if (lane < nvalid) {
            int e = perm[base + lane];
            srcv = ei[e];
            dstv = ei[EE + e];
            iv = inv[r * NN + dstv];
        }
        sDst[wave][lane] = dstv;
        sSrc[wave][lane] = srcv;
        sInv[wave][lane] = iv;
    }
    asm volatile("s_wait_dscnt 0x0" ::: "memory");
    // async-stage 16 gathered source rows into LDS
    for (int i = 0; i < 16; ++i) {
        int s = sSrc[wave][i];
        const float* gp = x + (size_t)s * DD + lane * 4;
        unsigned loff = (unsigned)(size_t)&lds[wave][i][lane * 4];
        asm volatile("global_load_async_to_lds_b128 %0, %1, off"
                     :: "v"(loff), "v"(gp) : "memory");
    }
    asm volatile("s_wait_asynccnt 0x0" ::: "memory");
    int hi = lane >> 4, mlo = lane & 15;
    for (int nt = 0; nt < 8; ++nt) {
        v8f acc = {};
        const float* Bp = Wp + (size_t)(r * 8 + nt) * 16 * 128;
        for (int k2 = 0; k2 < 16; ++k2) {
            float4 bb = *(const float4*)(Bp + k2 * 128 + lane * 4);
            const float* arow = &lds[wave][mlo][8 * k2 + 2 * hi];
            v2f a0; a0.x = arow[0]; a0.y = arow[1];
            v2f b0; b0.x = bb.x; b0.y = bb.y;
            acc = __builtin_amdgcn_wmma_f32_16x16x4_f32(false, a0, false, b0,
                                                        (short)0, acc, false, false);
            v2f a1; a1.x = arow[4]; a1.y = arow[5];
            v2f b1; b1.x = bb.z; b1.y = bb.w;
            acc = __builtin_amdgcn_wmma_f32_16x16x4_f32(false, a1, false, b1,
                                                        (short)0, acc, false, false);
        }
        for (int v = 0; v < 8; ++v) {
            int m = v + 8 * hi;
            int dd = sDst[wave][m];
            if (dd >= 0)
                atomicAdd(&h[dd * DD + nt * 16 + mlo], acc[v] * sInv[wave][m]);
        }
    }
}

// ---------------------------------------------------------------- LN + relu + residual
__global__ void __launch_bounds__(128) k_ln(const float* __restrict__ h,
                                            const float* __restrict__ xin,
                                            const float* __restrict__ gamma,
                                            const float* __restrict__ beta,
                                            float* __restrict__ xout) {
    int wave = threadIdx.x >> 5, lane = threadIdx.x & 31;
    int row = blockIdx.x * 4 + wave;
    float4 hv = *(const float4*)(h + row * DD + lane * 4);
    float s = hv.x + hv.y + hv.z + hv.w;
    float s2 = hv.x * hv.x + hv.y * hv.y + hv.z * hv.z + hv.w * hv.w;
    for (int m = 16; m >= 1; m >>= 1) {
        s += __shfl_xor(s, m);
        s2 += __shfl_xor(s2, m);
    }
    float mu = s * (1.f / 128.f);
    float var = s2 * (1.f / 128.f) - mu * mu;
    float rs = rsqrtf(var + 1e-5f);
    float4 xv = *(const float4*)(xin + row * DD + lane * 4);
    float g0 = gamma[lane * 4 + 0], g1 = gamma[lane * 4 + 1], g2 = gamma[lane * 4 + 2], g3 = gamma[lane * 4 + 3];
    float b0 = beta[lane * 4 + 0], b1 = beta[lane * 4 + 1], b2 = beta[lane * 4 + 2], b3 = beta[lane * 4 + 3];
    float4 o;
    o.x = fmaxf((hv.x - mu) * rs * g0 + b0, 0.f) + xv.x;
    o.y = fmaxf((hv.y - mu) * rs * g1 + b1, 0.f) + xv.y;
    o.z = fmaxf((hv.z - mu) * rs * g2 + b2, 0.f) + xv.z;
    o.w = fmaxf((hv.w - mu) * rs * g3 + b3, 0.f) + xv.w;
    *(float4*)(xout + row * DD + lane * 4) = o;
}

// ---------------------------------------------------------------- fused attention + score
__global__ void __launch_bounds__(128) k_attn(const float* __restrict__ x,
                                              const int* __restrict__ drugIdx,
                                              const int* __restrict__ phenoIdx,
                                              const unsigned char* __restrict__ mask,
                                              const float* __restrict__ Wqkv,
                                              const float* __restrict__ bqkv,
                                              const float* __restrict__ outW,
                                              const float* __restrict__ outB,
                                              const float* __restrict__ scoreW,
                                              const float* __restrict__ scoreB,
                                              float* __restrict__ out) {
    __shared__ float ph[PP][DD];
    __shared__ float dr[DD];
    __shared__ float q[DD];
    __shared__ float u[HH][DD];
    __shared__ float qb[HH];
    __shared__ float sc[HH][PP];
    __shared__ float t[HH][DD];
    __shared__ float ctx[DD];
    __shared__ float red[128];
    __shared__ int pidx[PP];
    int b = blockIdx.x, tid = threadIdx.x;
    if (tid < PP) pidx[tid] = phenoIdx[b * PP + tid];
    int di = drugIdx[b];
    dr[tid] = x[di * DD + tid];
    __syncthreads();
    for (int i = tid; i < PP * DD; i += 128) {
        int p = i >> 7, d = i & 127;
        ph[p][d] = x[pidx[p] * DD + d];
    }
    __syncthreads();
    // q = drug @ wq^T + bq
    {
        const float* wq = Wqkv + tid * DD;
        float acc = bqkv[tid];
        for (int d = 0; d < DD; ++d) acc += dr[d] * wq[d];
        q[tid] = acc;
    }
    __syncthreads();
    // u[h][d] = sum_j q[h*32+j] * wk[h*32+j][d]   ; qb[h] = q[h]·bk[h]
    for (int hh = 0; hh < HH; ++hh) {
        float acc = 0.f;
        for (int j = 0; j < DHH; ++j)
            acc += q[hh * DHH + j] * Wqkv[(DD + hh * DHH + j) * DD + tid];
        u[hh][tid] = acc;
    }
    if (tid < HH) {
        float acc = 0.f;
        for (int j = 0; j < DHH; ++j) acc += q[tid * DHH + j] * bqkv[DD + tid * DHH + j];
        qb[tid] = acc;
    }
    __syncthreads();
    // scores s[h][p] = (ph[p]·u[h] + qb[h]) / sqrt(dh)
    for (int i = tid; i < HH * PP; i += 128) {
        int hh = i >> 6, p = i & 63;
        float acc = qb[hh];
        for (int d = 0; d < DD; ++d) acc += ph[p][d] * u[hh][d];
        acc *= 0.17677669529663687f; // 1/sqrt(32)
        if (mask[b * PP + p]) acc = -1e9f;
        sc[hh][p] = acc;
    }
    __syncthreads();
    if (tid < HH) {
        float mx = -3.4e38f;
        for (int p = 0; p < PP; ++p) mx = fmaxf(mx, sc[tid][p]);
        float sum = 0.f;
        for (int p = 0; p < PP; ++p) { float e = expf(sc[tid][p] - mx); sc[tid][p] = e; sum += e; }
        float rr = 1.f / sum;
        for (int p = 0; p < PP; ++p) sc[tid][p] *= rr;
    }
    __syncthreads();
    // t[h][d] = sum_p attn[h][p] * ph[p][d]
    for (int i = tid; i < HH * DD; i += 128) {
        int hh = i >> 7, d = i & 127;
        float acc = 0.f;
        for (int p = 0; p < PP; ++p) acc += sc[hh][p] * ph[p][d];
        t[hh][d] = acc;
    }
    __syncthreads();
    // ctx[j] = t[h(j)]·wv[j] + bv[j]   (sum attn == 1 so bias folds directly)
    {
        int j = tid, hh = j >> 5;
        const float* wv = Wqkv + (2 * DD + j) * DD;
        float acc = bqkv[2 * DD + j];
        for (int d = 0; d < DD; ++d) acc += t[hh][d] * wv[d];
        ctx[j] = acc;
    }
    __syncthreads();
    // attn_out[o]·score_w[o], block reduce
    {
        const float* wr = outW + tid * DD;
        float acc = outB[tid];
        for (int j = 0; j < DD; ++j) acc += ctx[j] * wr[j];
        red[tid] = acc * scoreW[tid];
    }
    __syncthreads();
    for (int s = 64; s > 0; s >>= 1) {
        if (tid < s) red[tid] += red[tid + s];
        __syncthreads();
    }
    if (tid == 0) out[b] = red[0] + scoreB[0];
}

extern "C" void kernel_launch(void* const* d_in, const int* in_sizes, int n_in,
                              void* d_out, int out_size, void* d_ws, size_t ws_size,
                              hipStream_t stream) {
    const int* edge_index = (const int*)d_in[0];
    const int* edge_type  = (const int*)d_in[1];
    const int* drug_idx   = (const int*)d_in[2];
    const int* pheno_idx  = (const int*)d_in[3];
    const unsigned char* pheno_mask = (const unsigned char*)d_in[4];
    const float* node_emb = (const float*)d_in[5];
    const float* basis0 = (const float*)d_in[6];
    const float* comp0  = (const float*)d_in[7];
    const float* root0  = (const float*)d_in[8];
    const float* cbias0 = (const float*)d_in[9];
    const float* gamma0 = (const float*)d_in[10];
    const float* beta0  = (const float*)d_in[11];
    const float* basis1 = (const float*)d_in[12];
    const float* comp1  = (const float*)d_in[13];
    const float* root1  = (const float*)d_in[14];
    const float* cbias1 = (const float*)d_in[15];
    const float* gamma1 = (const float*)d_in[16];
    const float* beta1  = (const float*)d_in[17];
    const float* in_proj_w = (const float*)d_in[18];
    const float* in_proj_b = (const float*)d_in[19];
    const float* out_w  = (const float*)d_in[20];
    const float* out_b  = (const float*)d_in[21];
    const float* score_w = (const float*)d_in[22];
    const float* score_b = (const float*)d_in[23];
    float* outp = (float*)d_out;

    char* p = (char*)d_ws;
    auto carve = [&](size_t bytes) -> void* {
        void* r = (void*)p;
        p += (bytes + 255) & ~(size_t)255;
        return r;
    };
    float* wsWp = (float*)carve((size_t)RR * DD * DD * 4);   // packed relation weights
    float* wsRp = (float*)carve((size_t)DD * DD * 4);        // packed root
    float* cnt  = (float*)carve((size_t)RR * NN * 4);
    int*   perm = (int*)carve((size_t)EE * 4);
    int*   meta = (int*)carve(512);
    float* bufA = (float*)carve((size_t)NN * DD * 4);
    float* bufB = (float*)carve((size_t)NN * DD * 4);

    hipMemsetAsync(meta, 0, 512, stream);
    hipMemsetAsync(cnt, 0, (size_t)RR * NN * 4, stream);

    const int EB = (EE + 255) / 256;
    k_type_count<<<EB, 256, 0, stream>>>(edge_type, meta);
    k_scan<<<1, 32, 0, stream>>>(meta);
    k_permute<<<EB, 256, 0, stream>>>(edge_type, meta, perm);
    k_seg_count<<<EB, 256, 0, stream>>>(edge_index, edge_type, cnt);
    k_inv<<<(RR * NN + 255) / 256, 256, 0, stream>>>(cnt);

    const int MAXT = EE / 16 + RR;              // upper bound on relation tiles
    const int SCATTER_BLOCKS = (MAXT + 3) / 4;  // 4 waves / block
    const int ROOT_BLOCKS = (NN / 16 + 3) / 4;
    const int LN_BLOCKS = NN / 4;

    // ----- layer 0
    k_make_Wp<<<(RR * 8 * 16 * 32 + 255) / 256, 256, 0, stream>>>(comp0, basis0, wsWp);
    k_pack_root<<<(8 * 16 * 32 + 255) / 256, 256, 0, stream>>>(root0, wsRp);
    k_root<<<ROOT_BLOCKS, 128, 0, stream>>>(node_emb, wsRp, cbias0, bufA);
    k_scatter<<<SCATTER_BLOCKS, 128, 0, stream>>>(edge_index, node_emb, wsWp, cnt, perm, meta, bufA);
    k_ln<<<LN_BLOCKS, 128, 0, stream>>>(bufA, node_emb, gamma0, beta0, bufB);

    // ----- layer 1
    k_make_Wp<<<(RR * 8 * 16 * 32 + 255) / 256, 256, 0, stream>>>(comp1, basis1, wsWp);
    k_pack_root<<<(8 * 16 * 32 + 255) / 256, 256, 0, stream>>>(root1, wsRp);
    k_root<<<ROOT_BLOCKS, 128, 0, stream>>>(bufB, wsRp, cbias1, bufA);
    k_scatter<<<SCATTER_BLOCKS, 128, 0, stream>>>(edge_index, bufB, wsWp, cnt, perm, meta, bufA);
    k_ln<<<LN_BLOCKS, 128, 0, stream>>>(bufA, bufB, gamma1, beta1, bufA); // in-place OK (row-local)

    // ----- attention + score
    k_attn<<<BB, 128, 0, stream>>>(bufA, drug_idx, pheno_idx, pheno_mask,
                                   in_proj_w, in_proj_b, out_w, out_b,
                                   score_w, score_b, outp);
}